// SSM_10075993276628
// MI455X (gfx1250) — compile-verified
//
#include <hip/hip_runtime.h>

typedef float v2f __attribute__((ext_vector_type(2)));
typedef float v8f __attribute__((ext_vector_type(8)));

#define BATCH   4
#define SEQ     2048
#define DIN     1024          // IN_FEATURES == DIM_INNER
#define RANK    64            // dt_rank
#define NS      16            // d_state
#define NBC     96            // dt_rank + 2*d_state
#define CHUNK   64
#define NCHUNK  (SEQ / CHUNK) // 32
#define MROWS   (BATCH * SEQ) // 8192

static __device__ __forceinline__ v8f wmma_f32_16x16x4(v2f a, v2f b, v8f c) {
  // D = A(16x4,f32) * B(4x16,f32) + C(16x16,f32)  -> v_wmma_f32_16x16x4_f32
  return __builtin_amdgcn_wmma_f32_16x16x4_f32(false, a, false, b, (short)0, c,
                                               false, false);
}

// ---------------------------------------------------------------------------
// GEMM1: deltaBC[8192,96] = X[8192,1024] @ W_deltaBC^T  (W is [96,1024])
// One wave computes a full 16x96 strip (6 N-tiles), K swept in steps of 4.
// ---------------------------------------------------------------------------
__global__ void __launch_bounds__(256) gemm_deltaBC(const float* __restrict__ X,
                                                    const float* __restrict__ W,
                                                    float* __restrict__ out) {
  const int lane    = threadIdx.x & 31;
  const int wave    = threadIdx.x >> 5;
  const int m0      = (blockIdx.x * 8 + wave) * 16;
  const int halfsel = lane >> 4;       // selects K pair {0,1} vs {2,3}
  const int lidx    = lane & 15;       // M row (A) / N col (B)
  const int arow    = m0 + lidx;

  v8f acc[6] = {};
  for (int k = 0; k < DIN; k += 4) {
    const int kk = k + 2 * halfsel;
    v2f a = *(const v2f*)(X + arow * DIN + kk);
#pragma unroll
    for (int nt = 0; nt < 6; ++nt) {
      const int col = nt * 16 + lidx;
      v2f b = *(const v2f*)(W + col * DIN + kk);   // B[k][n] = W[n][k]
      acc[nt] = wmma_f32_16x16x4(a, b, acc[nt]);
    }
  }
#pragma unroll
  for (int nt = 0; nt < 6; ++nt)
#pragma unroll
    for (int r = 0; r < 8; ++r) {
      const int m = m0 + r + 8 * halfsel;
      out[m * NBC + nt * 16 + lidx] = acc[nt][r];
    }
}

// ---------------------------------------------------------------------------
// GEMM2: delta[8192,1024] = softplus(deltaBC[:, :64] @ W_dt^T + b_dt)
// One wave per 16x16 tile; K=64 fully unrolled (16 back-to-back WMMAs).
// ---------------------------------------------------------------------------
__global__ void __launch_bounds__(256) gemm_delta(const float* __restrict__ dBC,
                                                  const float* __restrict__ Wdt,
                                                  const float* __restrict__ bdt,
                                                  float* __restrict__ delta) {
  const int lane    = threadIdx.x & 31;
  const int wave    = threadIdx.x >> 5;
  const int wid     = blockIdx.x * 8 + wave;     // 0..32767
  const int mt      = wid >> 6;
  const int nt      = wid & 63;
  const int halfsel = lane >> 4;
  const int lidx    = lane & 15;
  const int m0 = mt * 16, n0 = nt * 16;
  const int arow = m0 + lidx, bcol = n0 + lidx;

  v8f acc = {};
#pragma unroll
  for (int k = 0; k < RANK; k += 4) {
    const int kk = k + 2 * halfsel;
    v2f a = *(const v2f*)(dBC + arow * NBC + kk);
    v2f b = *(const v2f*)(Wdt + bcol * RANK + kk);
    acc = wmma_f32_16x16x4(a, b, acc);
  }
  const float bias = bdt[bcol];
#pragma unroll
  for (int r = 0; r < 8; ++r) {
    const int m = m0 + r + 8 * halfsel;
    float v  = acc[r] + bias;
    float sp = (v > 20.0f) ? v : __logf(1.0f + __expf(v));
    delta[m * DIN + bcol] = sp;
  }
}

// ---------------------------------------------------------------------------
// Scan phase 1: per-chunk local scan with h0 = 0; emit carries (prod a, h_end)
// One thread owns one channel d; 16 states in registers; B-vecs staged in LDS.
// ---------------------------------------------------------------------------
__global__ void __launch_bounds__(256) scan_local(const float* __restrict__ delta,
                                                  const float* __restrict__ dBC,
                                                  const float* __restrict__ X,
                                                  const float* __restrict__ Alog,
                                                  float* __restrict__ Aprod,
                                                  float* __restrict__ Hend) {
  __shared__ float Bs[CHUNK][NS];
  const int tid = threadIdx.x;
  const int d   = blockIdx.x * 256 + tid;
  const int c   = blockIdx.y;
  const int b   = blockIdx.z;
  const int t0  = c * CHUNK;

  for (int i = tid; i < CHUNK * NS; i += 256) {
    const int t = i >> 4, s = i & 15;
    Bs[t][s] = dBC[(size_t)(b * SEQ + t0 + t) * NBC + RANK + s];
  }
  __syncthreads();

  float negA[NS];
#pragma unroll
  for (int s = 0; s < NS; ++s) negA[s] = -__expf(Alog[d * NS + s]);

  float h[NS], p[NS];
#pragma unroll
  for (int s = 0; s < NS; ++s) { h[s] = 0.0f; p[s] = 1.0f; }

  const size_t base = (size_t)(b * SEQ + t0) * DIN + d;
  for (int t = 0; t < CHUNK; ++t) {
    __builtin_prefetch(delta + base + (size_t)(t + 8) * DIN, 0, 0);
    __builtin_prefetch(X     + base + (size_t)(t + 8) * DIN, 0, 0);
    const float dv = delta[base + (size_t)t * DIN];
    const float xv = X[base + (size_t)t * DIN];
    const float dx = dv * xv;
#pragma unroll
    for (int s = 0; s < NS; ++s) {
      const float a = __expf(dv * negA[s]);
      h[s] = fmaf(a, h[s], dx * Bs[t][s]);
      p[s] *= a;
    }
  }
  const size_t co = ((size_t)((b * NCHUNK + c) * DIN) + d) * NS;
#pragma unroll
  for (int s = 0; s < NS; ++s) { Aprod[co + s] = p[s]; Hend[co + s] = h[s]; }
}

// ---------------------------------------------------------------------------
// Scan phase 2: sequential combine over the 32 chunk carries per (b,d,s),
// writing the incoming prefix state for each chunk.
// ---------------------------------------------------------------------------
__global__ void __launch_bounds__(256) scan_combine(const float* __restrict__ Aprod,
                                                    const float* __restrict__ Hend,
                                                    float* __restrict__ Hin) {
  const int idx = blockIdx.x * 256 + threadIdx.x;  // over BATCH*DIN*NS = 65536
  const int b   = idx / (DIN * NS);
  const int ds  = idx % (DIN * NS);
  float hp = 0.0f;
  for (int c = 0; c < NCHUNK; ++c) {
    const size_t o = (size_t)(b * NCHUNK + c) * (DIN * NS) + ds;
    Hin[o] = hp;
    hp = fmaf(Aprod[o], hp, Hend[o]);
  }
}

// ---------------------------------------------------------------------------
// Scan phase 3: re-scan each chunk seeded with Hin, contract with C, add D*x.
// ---------------------------------------------------------------------------
__global__ void __launch_bounds__(256) scan_final(const float* __restrict__ delta,
                                                  const float* __restrict__ dBC,
                                                  const float* __restrict__ X,
                                                  const float* __restrict__ Alog,
                                                  const float* __restrict__ Dp,
                                                  const float* __restrict__ Hin,
                                                  float* __restrict__ Y) {
  __shared__ float Bs[CHUNK][NS];
  __shared__ float Cs[CHUNK][NS];
  const int tid = threadIdx.x;
  const int d   = blockIdx.x * 256 + tid;
  const int c   = blockIdx.y;
  const int b   = blockIdx.z;
  const int t0  = c * CHUNK;

  for (int i = tid; i < CHUNK * NS; i += 256) {
    const int t = i >> 4, s = i & 15;
    const size_t row = (size_t)(b * SEQ + t0 + t) * NBC;
    Bs[t][s] = dBC[row + RANK + s];
    Cs[t][s] = dBC[row + RANK + NS + s];
  }
  __syncthreads();

  float negA[NS];
#pragma unroll
  for (int s = 0; s < NS; ++s) negA[s] = -__expf(Alog[d * NS + s]);

  float h[NS];
  const size_t co = ((size_t)((b * NCHUNK + c) * DIN) + d) * NS;
#pragma unroll
  for (int s = 0; s < NS; ++s) h[s] = Hin[co + s];

  const float dscale = Dp[d];
  const size_t base = (size_t)(b * SEQ + t0) * DIN + d;
  for (int t = 0; t < CHUNK; ++t) {
    __builtin_prefetch(delta + base + (size_t)(t + 8) * DIN, 0, 0);
    __builtin_prefetch(X     + base + (size_t)(t + 8) * DIN, 0, 0);
    const float dv = delta[base + (size_t)t * DIN];
    const float xv = X[base + (size_t)t * DIN];
    const float dx = dv * xv;
    float acc = 0.0f;
#pragma unroll
    for (int s = 0; s < NS; ++s) {
      const float a = __expf(dv * negA[s]);
      h[s] = fmaf(a, h[s], dx * Bs[t][s]);
      acc  = fmaf(h[s], Cs[t][s], acc);
    }
    Y[base + (size_t)t * DIN] = fmaf(dscale, xv, acc);
  }
}

// ---------------------------------------------------------------------------
extern "C" void kernel_launch(void* const* d_in, const int* in_sizes, int n_in,
                              void* d_out, int out_size, void* d_ws, size_t ws_size,
                              hipStream_t stream) {
  const float* x    = (const float*)d_in[0];  // [4,2048,1024]
  const float* WdBC = (const float*)d_in[1];  // [96,1024]
  const float* Wdt  = (const float*)d_in[2];  // [1024,64]
  const float* bdt  = (const float*)d_in[3];  // [1024]
  const float* Alog = (const float*)d_in[4];  // [1024,16]
  const float* Dp   = (const float*)d_in[5];  // [1024]
  float* y  = (float*)d_out;                  // [4,2048,1024]
  float* ws = (float*)d_ws;

  float* dBC   = ws;                                        //  786432 f
  float* delta = dBC   + (size_t)MROWS * NBC;               // 8388608 f
  float* Aprod = delta + (size_t)MROWS * DIN;               // 2097152 f
  float* Hend  = Aprod + (size_t)BATCH * NCHUNK * DIN * NS; // 2097152 f
  float* Hin   = Hend  + (size_t)BATCH * NCHUNK * DIN * NS; // 2097152 f

  gemm_deltaBC<<<dim3(MROWS / 16 / 8), dim3(256), 0, stream>>>(x, WdBC, dBC);
  gemm_delta  <<<dim3((MROWS / 16) * (DIN / 16) / 8), dim3(256), 0, stream>>>(dBC, Wdt, bdt, delta);
  scan_local  <<<dim3(DIN / 256, NCHUNK, BATCH), dim3(256), 0, stream>>>(delta, dBC, x, Alog, Aprod, Hend);
  scan_combine<<<dim3(BATCH * DIN * NS / 256), dim3(256), 0, stream>>>(Aprod, Hend, Hin);
  scan_final  <<<dim3(DIN / 256, NCHUNK, BATCH), dim3(256), 0, stream>>>(delta, dBC, x, Alog, Dp, Hin, y);
}